// EarModel_65472481460466
// MI455X (gfx1250) — compile-verified
//
#include <hip/hip_runtime.h>
#include <math.h>

// ---------------- problem constants ----------------
#define FS_F   24000.0f
#define T_LEN  48000
#define NCH    32
#define TILE   256                       // 16 samples/block x 16 blocks per WMMA tile
#define NT     ((T_LEN + TILE - 1) / TILE)
#define SMALLF 1e-30f
#define PI_F   3.14159265358979f

typedef float v2f __attribute__((ext_vector_type(2)));
typedef float v8f __attribute__((ext_vector_type(8)));

// ---------------- workspace layout (floats) ----------------
#define OFF_SIGF 0
#define OFF_CTRL ((size_t)OFF_SIGF + 2ull * T_LEN)
#define OFF_ENV  ((size_t)OFF_CTRL + 2ull * NCH * T_LEN)
#define OFF_BM   ((size_t)OFF_ENV  + 2ull * NCH * T_LEN)
#define OFF_DB   ((size_t)OFF_BM   + 2ull * NCH * T_LEN)
#define OFF_BM2  ((size_t)OFF_DB   + 2ull * NCH * T_LEN)
#define OFF_PRM  ((size_t)OFF_BM2  + 2ull * NCH * T_LEN)
// param block internal layout (offsets relative to P)
#define L_CF     0     // cf[32]
#define L_X      32    // aohc,bwmin,knee,cr,aihc  (5 x 32) for "x" ear
#define L_Y      192   // same for "y" ear
#define L_BW1    352   // bandwidth_1[32]
#define L_BWCTRL 384   // control bw duplicated for both signals [2][32]
#define L_BWADJ  448   // adjusted bw [2][32]
#define L_CSUM   512   // sum(ctrl_env^2) [2][32]
#define L_ESUM   576   // sum(env^2)      [2][32]
#define L_CORR   640   // int delay per channel [32]
#define PRM_SZ   672

// ---------------- helpers ----------------
__device__ inline void loss5(float loss, int c, float& aohc, float& bwmin,
                             float& knee, float& crout, float& aihc) {
  float cr = 1.25f + 2.25f * (float)c / 31.f;
  float th = 1.25f * 70.f * (1.f - 1.f / cr);
  float ao, ai;
  if (loss < th) { ao = 0.8f * loss; ai = 0.2f * loss; }
  else           { ao = 0.8f * th;   ai = 0.2f * th + (loss - th); }
  float qq = ao * 0.02f;
  float q2 = qq * qq;
  float q6 = q2 * q2 * q2;
  bwmin = 1.f + qq + 2.f * q6;
  knee  = ao + 30.f;
  float up = 30.f + 70.f / cr;
  crout = (100.f - knee) / (up + ao - knee);
  aohc = ao; aihc = ai;
}

__device__ inline float gauss1(unsigned long long key) {
  unsigned long long z = key + 0x9E3779B97F4A7C15ull;   // splitmix64
  z = (z ^ (z >> 30)) * 0xBF58476D1CE4E5B9ull;
  z = (z ^ (z >> 27)) * 0x94D049BB133111EBull;
  z ^= z >> 31;
  unsigned int u1 = (unsigned int)z, u2 = (unsigned int)(z >> 32);
  float f1 = ((u1 >> 8) + 1u) * (1.f / 16777216.f);
  float f2 = (u2 >> 8) * (1.f / 16777216.f);
  float r = sqrtf(-2.f * __logf(f1));
  float sn, cn; __sincosf(6.28318530718f * f2, &sn, &cn);
  return r * cn;
}

// ---------------- 1) Butterworth prefilter (2 serial chains) ----------------
__global__ void k_prefilter(const float* __restrict__ ref,
                            const float* __restrict__ proc,
                            float* __restrict__ out) {
  int lane = threadIdx.x;
  if (lane >= 2) return;
  const float* x = lane ? proc : ref;
  float K1 = tanf(PI_F * 5000.f / FS_F);
  float b0l = K1 / (1.f + K1), b1l = b0l, a1l = (K1 - 1.f) / (1.f + K1);
  float K2 = tanf(PI_F * 350.f / FS_F);
  float n = 1.f / (1.f + 1.41421356f * K2 + K2 * K2);
  float b0h = n, b1h = -2.f * n, b2h = n;
  float a1h = 2.f * (K2 * K2 - 1.f) * n, a2h = (1.f - 1.41421356f * K2 + K2 * K2) * n;
  float z1 = 0.f, w1 = 0.f, w2 = 0.f;
  float* o = out + (size_t)lane * T_LEN;
  for (int t = 0; t < T_LEN; ++t) {
    float xt = x[t];
    float y1 = b0l * xt + z1;  z1 = b1l * xt - a1l * y1;           // DF2T 1st order LP
    float y2 = b0h * y1 + w1;                                       // DF2T 2nd order HP
    w1 = b1h * y1 + w2 - a1h * y2;
    w2 = b2h * y1 - a2h * y2;
    o[t] = y2;
  }
}

// ---------------- 2) center freqs + loss parameters ----------------
__global__ void k_params(const float* __restrict__ hl, const int* __restrict__ eqp,
                         float* __restrict__ P) {
  int c = threadIdx.x;
  if (c >= NCH) return;
  const float q = 9.26449f * 24.7f;
  auto cf_of = [&](int k) -> float {
    if (k == 0) return 8000.f;
    return -q + __expf((float)k * (__logf(80.f + q) - __logf(8000.f + q)) * (1.f / 31.f)) * (8000.f + q);
  };
  float cfc = cf_of(31 - c);
  float cf0 = cf_of(31);
  P[L_CF + c] = cfc;

  float f_v[8]  = {cf0, 250.f, 500.f, 1000.f, 2000.f, 4000.f, 6000.f, 8000.f};
  float hl_e[8] = {hl[0], hl[0], hl[1], hl[2], hl[3], hl[4], hl[5], hl[5]};
  float loss;
  if (cfc <= f_v[0]) loss = hl_e[0];
  else if (cfc >= f_v[7]) loss = hl_e[7];
  else {
    loss = hl_e[7];
    for (int i = 0; i < 7; ++i) {
      if (cfc <= f_v[i + 1]) {
        float tf = (cfc - f_v[i]) / (f_v[i + 1] - f_v[i]);
        loss = hl_e[i] + tf * (hl_e[i + 1] - hl_e[i]);
        break;
      }
    }
  }
  loss = fmaxf(loss, 0.f);

  float ao, bwm, kn, co, ai;
  loss5(loss, c, ao, bwm, kn, co, ai);                       // y ear
  P[L_Y + c] = ao; P[L_Y + 32 + c] = bwm; P[L_Y + 64 + c] = kn;
  P[L_Y + 96 + c] = co; P[L_Y + 128 + c] = ai;

  float loss_x = (*eqp == 0) ? 0.f : loss;                   // x ear
  loss5(loss_x, c, ao, bwm, kn, co, ai);
  P[L_X + c] = ao; P[L_X + 32 + c] = bwm; P[L_X + 64 + c] = kn;
  P[L_X + 96 + c] = co; P[L_X + 128 + c] = ai;

  loss5(100.f, c, ao, bwm, kn, co, ai);                      // bandwidth_1 (hl = 100)
  P[L_BW1 + c] = bwm;
  P[L_BWCTRL + c] = bwm;
  P[L_BWCTRL + 32 + c] = bwm;
}

// ---------------- 3) gammatone bank via WMMA blocked parallel scan ----------------
// Denominator (1 - a z^-1)^4 => FIR [1,4a,4a^2] then FOUR identical one-pole passes.
// Each pass over a 256-sample tile: Y(16x16) = L(a)(16x16, lower-tri Toeplitz) @ X(16x16),
// computed with 4 chained V_WMMA_F32_16X16X4_F32, then a 16-step serial carry fixup.
// One wave = one complex component (re or im) of one (signal, channel) chain.
__global__ __launch_bounds__(64) void k_gammatone(
    const float* __restrict__ sig, const float* __restrict__ P,
    const float* __restrict__ bw, float* __restrict__ env_out,
    float* __restrict__ bm_out, float* __restrict__ sumsq, int write_bm) {
  const int blk = blockIdx.x;            // 0..63 = (signal, channel)
  const int s = blk >> 5, c = blk & 31;
  const int tid = threadIdx.x;           // 0..63: wave0=real, wave1=imag
  const int wv = tid >> 5;
  const int lane = tid & 31;
  const int lo = lane & 15;
  const int hi = lane >> 4;

  __shared__ float xm_buf[2][TILE + 2];
  __shared__ float u_buf[2][TILE];
  __shared__ float rowb[2][16];
  __shared__ float carb[2][16];
  __shared__ float carry[2][4];
  __shared__ float red[64];

  const float tpt = 2.f * PI_F / FS_F;
  const float cfc = P[L_CF + c];
  const float erb = 24.7f + cfc * (1.f / 9.26449f);
  const float aa  = __expf(-tpt * 1.019f * bw[s * NCH + c] * erb);
  const float a1  = 4.f * aa;
  const float aa2 = aa * aa;
  const float a5  = 4.f * aa2;
  const float a2m = -6.f * aa2;
  const float a3m = 4.f * aa2 * aa;
  const float a4m = -aa2 * aa2;
  const float gain = 2.f * (1.f - a1 - a2m - a3m - a4m) / (1.f + a1 + a5);
  const float wph  = tpt * cfc;

  // A operand: L(a) split into four 16x4 K-chunks (wave32 f32 A layout:
  // lanes 0-15 row M=lo with K={4k,4k+1}; lanes 16-31 K={4k+2,4k+3}).
  v2f Amat[4];
  for (int k = 0; k < 4; ++k) {
    int j0 = 4 * k + 2 * hi;
    Amat[k].x = (lo >= j0)     ? powf(aa, (float)(lo - j0))     : 0.f;
    Amat[k].y = (lo >= j0 + 1) ? powf(aa, (float)(lo - j0 - 1)) : 0.f;
  }
  float apow[8];
  for (int r = 0; r < 8; ++r) apow[r] = powf(aa, (float)(r + 8 * hi + 1));
  const float a16 = powf(aa, 16.f);

  if (lane < 4) carry[wv][lane] = 0.f;
  float ssum = 0.f;
  __syncthreads();

  const float* sg = sig + (size_t)s * T_LEN;

  for (int tile = 0; tile < NT; ++tile) {
    const int t0 = tile * TILE;
    __builtin_prefetch(sg + ((t0 + TILE < T_LEN) ? (t0 + TILE) : 0), 0, 1);

    // stage demodulated signal with 2-sample FIR halo
    for (int j = lane; j < TILE + 2; j += 32) {
      int t = t0 - 2 + j;
      float v = 0.f;
      if (t >= 0 && t < T_LEN) {
        float sn, cn; __sincosf(wph * (float)t, &sn, &cn);
        v = sg[t] * (wv ? sn : cn);
      }
      xm_buf[wv][j] = v;
    }
    __syncthreads();
    // numerator FIR [1, 4a, 4a^2]
    for (int lt = lane; lt < TILE; lt += 32)
      u_buf[wv][lt] = xm_buf[wv][lt + 2] + a1 * xm_buf[wv][lt + 1] + a5 * xm_buf[wv][lt];
    __syncthreads();

    // four cascaded one-pole passes
    for (int p = 0; p < 4; ++p) {
      v2f B0, B1, B2, B3;
      {
        const float* u = u_buf[wv];
        int r0 = 2 * hi, n16 = 16 * lo;   // B layout: lane = column N, VGPR pair = K rows
        B0.x = u[(0  + r0) + n16];  B0.y = u[(1  + r0) + n16];
        B1.x = u[(4  + r0) + n16];  B1.y = u[(5  + r0) + n16];
        B2.x = u[(8  + r0) + n16];  B2.y = u[(9  + r0) + n16];
        B3.x = u[(12 + r0) + n16];  B3.y = u[(13 + r0) + n16];
      }
      v8f acc = {0.f, 0.f, 0.f, 0.f, 0.f, 0.f, 0.f, 0.f};
      acc = __builtin_amdgcn_wmma_f32_16x16x4_f32(false, Amat[0], false, B0, (short)0, acc, false, false);
      acc = __builtin_amdgcn_wmma_f32_16x16x4_f32(false, Amat[1], false, B1, (short)0, acc, false, false);
      acc = __builtin_amdgcn_wmma_f32_16x16x4_f32(false, Amat[2], false, B2, (short)0, acc, false, false);
      acc = __builtin_amdgcn_wmma_f32_16x16x4_f32(false, Amat[3], false, B3, (short)0, acc, false, false);

      if (hi) rowb[wv][lo] = acc[7];   // raw last row (M=15) of each block
      __syncthreads();
      if (lane == 0) {                 // serial exclusive scan of cross-block carries
        float cc = carry[wv][p];
        for (int b = 0; b < 16; ++b) {
          carb[wv][b] = cc;
          cc = rowb[wv][b] + a16 * cc;
        }
        carry[wv][p] = cc;
      }
      __syncthreads();
      float cb = carb[wv][lo];
      for (int r = 0; r < 8; ++r)
        u_buf[wv][(r + 8 * hi) + 16 * lo] = acc[r] + apow[r] * cb;
      __syncthreads();
    }

    // env / bm from real (wave0) + imag (wave1) chains
    for (int lt = tid; lt < TILE; lt += 64) {
      int t = t0 + lt;
      if (t < T_LEN) {
        float ur = u_buf[0][lt], ui = u_buf[1][lt];
        float e = gain * sqrtf(ur * ur + ui * ui);
        env_out[(size_t)blk * T_LEN + t] = e;
        ssum += e * e;
        if (write_bm) {
          float sn, cn; __sincosf(wph * (float)t, &sn, &cn);
          bm_out[(size_t)blk * T_LEN + t] = gain * (ur * cn + ui * sn);
        }
      }
    }
    __syncthreads();
  }

  red[tid] = ssum;
  __syncthreads();
  for (int st = 32; st > 0; st >>= 1) {
    if (tid < st) red[tid] += red[tid + st];
    __syncthreads();
  }
  if (tid == 0) sumsq[blk] = red[0];
}

// ---------------- 4) bandwidth adjust ----------------
__global__ void k_bw_adjust(float* __restrict__ P, const int* __restrict__ lv1p) {
  int l = threadIdx.x;
  if (l >= 64) return;
  int s = l >> 5, c = l & 31;
  float lv1 = (float)(*lv1p);
  float cavg = sqrtf(P[L_CSUM + l] / (float)T_LEN);
  float cdb = 20.f * log10f(cavg + SMALLF) + lv1;
  float frac = fminf(fmaxf((cdb - 50.f) / 50.f, 0.f), 1.f);
  float bwmin = P[(s ? L_Y : L_X) + 32 + c];
  P[L_BWADJ + l] = bwmin + frac * (P[L_BW1 + c] - bwmin);
}

// ---------------- 5) compress + SL + IHC adaptation (64 lane-parallel chains) ----------------
__global__ __launch_bounds__(64) void k_compress_ihc(
    const float* __restrict__ P, const float* __restrict__ ctrl,
    const float* __restrict__ env, const float* __restrict__ bm,
    float* __restrict__ db, float* __restrict__ bm2, const int* __restrict__ lv1p) {
  int l = threadIdx.x;
  if (l >= 64) return;
  int s = l >> 5, c = l & 31;
  const float* par = P + (s ? L_Y : L_X);
  float aohc = par[c], knee = par[64 + c], crr = par[96 + c], aihc = par[128 + c];
  float lv1 = (float)(*lv1p);

  float K = tanf(PI_F * 800.f / FS_F);                 // 800 Hz one-pole LP
  float b0 = K / (1.f + K), b1 = b0, a1f = (K - 1.f) / (1.f + K);

  float r1 = 0.5f, r2 = 0.25f, r3 = 0.25f;             // IHC adapt (delta = 2)
  float Ts = 1.f / FS_F;
  float c1 = 0.002f * (r1 + r2) / (r1 * r2), c2 = 0.06f / r3;
  float a11 = r1 + r2 + r1 * r2 * (c1 / Ts);
  float a12 = -r1, a21 = -r3;
  float a22 = r2 + r3 + r2 * r3 * (c2 / Ts);
  float den = 1.f / (a11 * a22 - a21 * a12);
  float r1c = r1 * r2 * (c1 / Ts), r2c = r2 * r3 * (c2 / Ts);
  float gslope = 1.f - 1.f / crr;

  float z = 0.f, v1 = 0.f, v2 = 0.f;
  const size_t off = (size_t)l * T_LEN;
  for (int t = 0; t < T_LEN; ++t) {
    float ctl = ctrl[off + t];
    float logenv = lv1 + 20.f * log10f(fmaxf(ctl, SMALLF));
    logenv = fminf(fmaxf(logenv, knee), 100.f);
    float gdb = (-aohc - (logenv - knee) * gslope) * 0.05f;
    float g0 = exp10f(gdb);
    float gf = b0 * g0 + z;
    z = b1 * g0 - a1f * gf;
    float xc = gf * env[off + t];
    float xb = gf * bm[off + t];
    float ysl = fmaxf(lv1 - aihc + 20.f * log10f(xc + SMALLF), 0.f);
    float bsl = ysl * xb / (xc + SMALLF);
    float b1v = ysl * r2 + r1c * v1;
    float b2v = r2c * v2;
    v1 = den * (a22 * b1v - a12 * b2v);
    v2 = den * (-a21 * b1v + a11 * b2v);
    float o = fmaxf(2.f * (ysl - v1), 0.f);            // (r1+r2+r3)*(y-v1)/r1 = 2*(y-v1)
    db[off + t] = o;
    bm2[off + t] = bsl * (o + 1e-5f) / (ysl + 1e-5f);
  }
}

// ---------------- 6) group-delay compensation parameters ----------------
__global__ void k_delay(float* __restrict__ P) {
  __shared__ float gds[NCH];
  int c = threadIdx.x;
  if (c >= NCH) return;
  float cfc = P[L_CF + c];
  float erb = 24.7f + cfc * (1.f / 9.26449f);
  float tpt = 2.f * PI_F / FS_F;
  float aa = __expf(-tpt * 1.019f * P[L_BWADJ + c] * erb);  // ref bw (signal 0)
  float a1 = 4.f * aa, aa2 = aa * aa;
  float a5 = 4.f * aa2, a2 = -6.f * aa2, a3 = 4.f * aa2 * aa, a4 = -aa2 * aa2;
  float num  = (a1 + 2.f * a5) / (1.f + a1 + a5);
  float dnum = (-a1 - 2.f * a2 - 3.f * a3 - 4.f * a4) / (1.f - a1 - a2 - a3 - a4);
  float gd = num - dnum;
  gds[c] = gd;
  __syncthreads();
  float mn = gds[0];
  for (int i = 1; i < NCH; ++i) mn = fminf(mn, gds[i]);
  __syncthreads();
  gd = rintf(gd - mn);
  gds[c] = gd;
  __syncthreads();
  float mx = gds[0];
  for (int i = 1; i < NCH; ++i) mx = fmaxf(mx, gds[i]);
  ((int*)(P + L_CORR))[c] = (int)(mx - gd);
}

// ---------------- 7) shift (group-delay comp) + noise injection ----------------
__global__ void k_output(const float* __restrict__ P, const float* __restrict__ db,
                         const float* __restrict__ bm2, float* __restrict__ out,
                         const int* __restrict__ lv1p) {
  int bid = blockIdx.x;                 // 0..127
  int arr = bid >> 5, c = bid & 31;
  const int* corr = (const int*)(P + L_CORR);
  int cr = corr[c];
  float lv1 = (float)(*lv1p);
  float ng = powf(10.f, (-10.f - lv1) * 0.05f);
  const float* src;
  int noise = 0;
  unsigned long long salt = 0;
  if (arr == 0)      { src = db  + (size_t)c * T_LEN; }
  else if (arr == 1) { src = bm2 + (size_t)c * T_LEN; noise = 1; salt = 0x1111111111111111ull; }
  else if (arr == 2) { src = db  + (size_t)(c + 32) * T_LEN; }
  else               { src = bm2 + (size_t)(c + 32) * T_LEN; noise = 1; salt = 0x2222222222222222ull; }
  float* dst = out + (size_t)arr * NCH * T_LEN + (size_t)c * T_LEN;
  for (int t = threadIdx.x; t < T_LEN; t += blockDim.x) {
    int ts = t - cr;
    float v = 0.f;
    if (ts >= 0) {
      v = src[ts];
      if (noise)
        v += ng * gauss1(salt ^ ((unsigned long long)c << 32) ^ (unsigned long long)(unsigned int)ts);
    }
    dst[t] = v;
  }
}

// ---------------- 8) RMS -> SL + FS scalar ----------------
__global__ void k_final(const float* __restrict__ P, float* __restrict__ out,
                        const int* __restrict__ lv1p) {
  int l = threadIdx.x;
  float lv1 = (float)(*lv1p);
  if (l < 64) {
    int s = l >> 5, c = l & 31;
    const float* par = P + (s ? L_Y : L_X);
    float aohc = par[c], knee = par[64 + c], cr = par[96 + c], aihc = par[128 + c];
    float avg  = sqrtf(P[L_ESUM + l] / (float)T_LEN);
    float cavg = sqrtf(P[L_CSUM + l] / (float)T_LEN);
    float cdb = fminf(fmaxf(lv1 + 20.f * log10f(cavg + SMALLF), knee), 100.f);
    float g = -aohc - (cdb - knee) * (1.f - 1.f / cr);
    out[(size_t)4 * NCH * T_LEN + l] =
        fmaxf(lv1 + 20.f * log10f(avg + SMALLF) + g - aihc, 0.f);
  }
  if (l == 0) out[(size_t)4 * NCH * T_LEN + 64] = FS_F;
}

// ---------------- host launcher ----------------
extern "C" void kernel_launch(void* const* d_in, const int* in_sizes, int n_in,
                              void* d_out, int out_size, void* d_ws, size_t ws_size,
                              hipStream_t stream) {
  (void)in_sizes; (void)n_in; (void)out_size; (void)ws_size;
  const float* ref  = (const float*)d_in[0];
  const float* proc = (const float*)d_in[1];
  const float* hl   = (const float*)d_in[2];
  const int*   eq   = (const int*)d_in[5];
  const int*   lv1  = (const int*)d_in[6];
  float* out = (float*)d_out;
  float* ws  = (float*)d_ws;

  float* sigf = ws + OFF_SIGF;
  float* ctrl = ws + OFF_CTRL;
  float* env  = ws + OFF_ENV;
  float* bm   = ws + OFF_BM;
  float* db   = ws + OFF_DB;
  float* bm2  = ws + OFF_BM2;
  float* P    = ws + OFF_PRM;

  k_prefilter<<<1, 32, 0, stream>>>(ref, proc, sigf);
  k_params<<<1, 32, 0, stream>>>(hl, eq, P);
  // control filterbank (bandwidth_1)
  k_gammatone<<<64, 64, 0, stream>>>(sigf, P, P + L_BWCTRL, ctrl, ctrl, P + L_CSUM, 0);
  k_bw_adjust<<<1, 64, 0, stream>>>(P, lv1);
  // main filterbank (adjusted bandwidth), env + basilar membrane
  k_gammatone<<<64, 64, 0, stream>>>(sigf, P, P + L_BWADJ, env, bm, P + L_ESUM, 1);
  k_compress_ihc<<<1, 64, 0, stream>>>(P, ctrl, env, bm, db, bm2, lv1);
  k_delay<<<1, 32, 0, stream>>>(P);
  k_output<<<128, 256, 0, stream>>>(P, db, bm2, out, lv1);
  k_final<<<1, 64, 0, stream>>>(P, out, lv1);
}